// Interaction_layer_6700148981881
// MI455X (gfx1250) — compile-verified
//
#include <hip/hip_runtime.h>
#include <cstdint>
#include <cstddef>

// Problem constants (from reference): B=16384, F=26, D=32, pairs = 26*25/2 = 325
#define NF     26
#define ND     32
#define NPAIR  325
#define ROWF4  (ND / 4)          // 8 float4 per row
#define INF4   (NF * ROWF4)      // 208 float4 of input per batch (3328 B)
#define OUTF4  (NPAIR * ROWF4)   // 2600 float4 of output per batch (41600 B)
#define BLOCK  256               // 8 wave32 waves

typedef float v4f __attribute__((ext_vector_type(4)));
typedef int   v4i __attribute__((ext_vector_type(4)));

#if defined(__AMDGCN__) && __has_builtin(__builtin_amdgcn_global_load_async_to_lds_b128)
#define HAS_ASYNC_LDS 1
#else
#define HAS_ASYNC_LDS 0
#endif

__global__ __launch_bounds__(BLOCK) void pairwise_interaction_kernel(
    const float* __restrict__ x, float* __restrict__ out)
{
    // Input tile for this batch element, staged once; each row reused 25x.
    __shared__ v4f            s_in[INF4];      // 3328 B
    __shared__ unsigned short s_pair[NPAIR];   // (i<<8)|j table, 650 B

    const int    tid = threadIdx.x;
    const size_t b   = blockIdx.x;

    const float* gsrc = x + b * (size_t)(NF * ND);

    // ---- Stage input 3328 B global -> LDS via CDNA5 async copy (ASYNCcnt) ----
#if HAS_ASYNC_LDS
    if (tid < INF4) {
        typedef __attribute__((address_space(1))) v4i gv4i;   // printed as __device__ int4*
        typedef __attribute__((address_space(3))) v4i lv4i;   // LDS int4*
        float* gflat = const_cast<float*>(gsrc) + (size_t)tid * 4;
        gv4i* gp = (gv4i*)(void*)gflat;
        lv4i* lp = (lv4i*)(void*)(&s_in[tid]);
        __builtin_amdgcn_global_load_async_to_lds_b128(gp, lp, 0, 0);
    }
#else
    if (tid < INF4) {
        s_in[tid] = ((const v4f*)gsrc)[tid];   // fallback: global_load_b128 + ds_store_b128
    }
#endif

    // ---- Build (i,j) pair table: row-major upper triangle, i outer, j inner ----
    for (int p = tid; p < NPAIR; p += BLOCK) {
        int i = 0, rem = p;
        while (rem >= (NF - 1 - i)) { rem -= (NF - 1 - i); ++i; }
        s_pair[p] = (unsigned short)((i << 8) | (i + 1 + rem));
    }

#if HAS_ASYNC_LDS
# if __has_builtin(__builtin_amdgcn_s_wait_asynccnt)
    __builtin_amdgcn_s_wait_asynccnt(0);
# else
    asm volatile("s_wait_asynccnt 0" ::: "memory");
# endif
#endif
    __syncthreads();

    // ---- Streaming compute: 2600 float4 per batch; NT b128 stores ----
    v4f* gout = (v4f*)(out + b * (size_t)(NPAIR * ND));

#pragma unroll
    for (int k = 0; k < OUTF4 / BLOCK; ++k) {        // 10 fully unrolled iterations
        const int w  = tid + k * BLOCK;
        const int p  = w >> 3;                       // pair index
        const int d4 = w & 7;                        // float4 chunk within D
        const int pr = s_pair[p];
        const v4f a  = s_in[((pr >> 8)   << 3) + d4];   // row i
        const v4f c  = s_in[((pr & 0xff) << 3) + d4];   // row j
        __builtin_nontemporal_store(a * c, gout + w);
    }
    {   // remainder: 2600 - 2560 = 40 float4
        const int w = tid + (OUTF4 / BLOCK) * BLOCK;
        if (w < OUTF4) {
            const int p  = w >> 3;
            const int d4 = w & 7;
            const int pr = s_pair[p];
            const v4f a  = s_in[((pr >> 8)   << 3) + d4];
            const v4f c  = s_in[((pr & 0xff) << 3) + d4];
            __builtin_nontemporal_store(a * c, gout + w);
        }
    }
}

extern "C" void kernel_launch(void* const* d_in, const int* in_sizes, int n_in,
                              void* d_out, int out_size, void* d_ws, size_t ws_size,
                              hipStream_t stream)
{
    (void)n_in; (void)out_size; (void)d_ws; (void)ws_size;
    const float* x   = (const float*)d_in[0];
    float*       out = (float*)d_out;
    const int    B   = in_sizes[0] / (NF * ND);   // 16384

    pairwise_interaction_kernel<<<dim3(B), dim3(BLOCK), 0, stream>>>(x, out);
}